// TGNMemory_54288386621730
// MI455X (gfx1250) — compile-verified
//
#include <hip/hip_runtime.h>
#include <math.h>

typedef float v2f __attribute__((ext_vector_type(2)));
typedef float v8f __attribute__((ext_vector_type(8)));

#define NODES_PER_BLK 32
#define THREADS 256
#define MEMD 100
#define DMSG 292
#define RAW 64
#define G3 300                 // 3 * MEMORY_DIM gate width
#define CHUNK_NODES 8          // nodes per FFN staging chunk
#define CHUNK_ROWS (CHUNK_NODES * RAW)   // 512 rows per chunk

__device__ __forceinline__ float gelu_exact(float x) {
    return 0.5f * x * (1.0f + erff(x * 0.70710678118654752440f));
}
__device__ __forceinline__ float sigmoidf_(float x) {
    return 1.0f / (1.0f + __expf(-x));
}

__global__ __launch_bounds__(THREADS) void tgn_update_kernel(
    const int*   __restrict__ n_id,
    const float* __restrict__ memory,   // [100000,100]
    const float* __restrict__ msg_d,    // [100000,64,20]
    const float* __restrict__ w1,  const float* __restrict__ b1,   // [10,20],[10]
    const float* __restrict__ w2,  const float* __restrict__ b2,   // [3,10],[3]
    const float* __restrict__ lg,  const float* __restrict__ lb,   // [3],[3]
    const float* __restrict__ lng, const float* __restrict__ lnb,  // [292],[292]
    const float* __restrict__ wih,                                 // [300,292]
    const float* __restrict__ whh,                                 // [300,100]
    float* __restrict__ out, int Ntot)
{
    __shared__ float s_x[NODES_PER_BLK * DMSG];     // msg_perm -> normalized x
    __shared__ float s_mem[NODES_PER_BLK * MEMD];   // raw gathered memory
    // union region: FFN staging (msg 10240 + h 5120 floats) OR gi/gh accum (2*9600)
    __shared__ float s_scr[NODES_PER_BLK * G3 * 2]; // 19200 floats = 76.8 KB
    __shared__ float s_w1[200], s_b1[10], s_w2[30], s_b2[3], s_lg[3], s_lb[3];
    __shared__ float s_lng[DMSG], s_lnb[DMSG];
    __shared__ float s_mu[NODES_PER_BLK], s_rstd[NODES_PER_BLK];
    __shared__ int   s_nid[NODES_PER_BLK];

    const int tid  = threadIdx.x;
    const int base = blockIdx.x * NODES_PER_BLK;
    const int wave = tid >> 5, lane = tid & 31;
    const int l16  = lane & 15;
    const int kh   = (lane >> 4) << 1;   // K sub-offset: 0 (lanes 0-15) / 2 (lanes 16-31)
    const int rhalf = (lane >> 4) << 3;  // C-fragment row offset: 0 / 8

    // ---- phase 0: indices + small weights into LDS ----
    if (tid < NODES_PER_BLK) {
        int g = base + tid; if (g > Ntot - 1) g = Ntot - 1;   // clamp tail
        s_nid[tid] = n_id[g];
    }
    for (int i = tid; i < 200; i += THREADS) s_w1[i] = w1[i];
    if (tid < 10) s_b1[tid] = b1[tid];
    if (tid < 30) s_w2[tid] = w2[tid];
    if (tid < 3)  { s_b2[tid] = b2[tid]; s_lg[tid] = lg[tid]; s_lb[tid] = lb[tid]; }
    for (int i = tid; i < DMSG; i += THREADS) { s_lng[i] = lng[i]; s_lnb[i] = lnb[i]; }
    __syncthreads();

    // ---- phase 1: gather memory rows (raw copy + into x[:,0:100]) ----
    for (int idx = tid; idx < NODES_PER_BLK * MEMD; idx += THREADS) {
        int node = idx / MEMD, k = idx % MEMD;
        float v = memory[(size_t)s_nid[node] * MEMD + k];
        s_mem[idx] = v;
        s_x[node * DMSG + k] = v;
    }

    // ---- phase 2: FeedForward, 8-node chunks; layer-1 (20->10) on WMMA ----
    float* s_msg = s_scr;                       // [512][20]
    float* s_h   = s_scr + CHUNK_ROWS * 20;     // [512][10]
    for (int c = 0; c < NODES_PER_BLK / CHUNK_NODES; ++c) {
        const int cbase = c * CHUNK_NODES;

        // stage msg rows for this chunk (coalesced)
        for (int idx = tid; idx < CHUNK_ROWS * 20; idx += THREADS) {
            int row = idx / 20, t = idx % 20;
            int node = cbase + (row >> 6), r = row & 63;
            s_msg[idx] = msg_d[((size_t)s_nid[node] * RAW + r) * 20 + t];
        }
        __syncthreads();

        // layer-1 WMMA: 32 M-tiles of 16 rows, K=20 (5 steps), N=10 padded to 16
        {
            const int jc = l16 < 10 ? l16 : 9;  // clamped B column
            for (int mt = wave; mt < CHUNK_ROWS / 16; mt += (THREADS >> 5)) {
                const int m0 = mt * 16;
                const float* ar = &s_msg[(m0 + l16) * 20];
                const float* br = &s_w1[jc * 20];
                v8f cacc = {};
                #pragma unroll
                for (int k0 = 0; k0 < 20; k0 += 4) {
                    int kk = k0 + kh;
                    v2f a = *(const v2f*)(ar + kk);
                    v2f b = *(const v2f*)(br + kk);
                    cacc = __builtin_amdgcn_wmma_f32_16x16x4_f32(
                            false, a, false, b, (short)0, cacc, false, false);
                }
                if (l16 < 10) {
                    float bias = s_b1[l16];
                    #pragma unroll
                    for (int v = 0; v < 8; ++v) {
                        int row = m0 + rhalf + v;
                        s_h[row * 10 + l16] = gelu_exact(cacc[v] + bias);
                    }
                }
            }
        }
        __syncthreads();

        // layer-2 (10->3) + LayerNorm(3) on VALU, write comp into s_x
        for (int rr = tid; rr < CHUNK_ROWS; rr += THREADS) {
            int node = cbase + (rr >> 6), r = rr & 63;
            const float* hv = &s_h[rr * 10];
            float y[3];
            #pragma unroll
            for (int o = 0; o < 3; ++o) {
                float a = s_b2[o];
                #pragma unroll
                for (int j = 0; j < 10; ++j) a += s_w2[o * 10 + j] * hv[j];
                y[o] = a;
            }
            float mu = (y[0] + y[1] + y[2]) * (1.0f / 3.0f);
            float d0 = y[0] - mu, d1 = y[1] - mu, d2 = y[2] - mu;
            float var = (d0 * d0 + d1 * d1 + d2 * d2) * (1.0f / 3.0f);
            float rs = rsqrtf(var + 1e-5f);
            float* xo = &s_x[node * DMSG + MEMD + r * 3];
            xo[0] = d0 * rs * s_lg[0] + s_lb[0];
            xo[1] = d1 * rs * s_lg[1] + s_lb[1];
            xo[2] = d2 * rs * s_lg[2] + s_lb[2];
        }
        __syncthreads();   // protect s_msg/s_h reuse next chunk
    }

    // ---- phase 3: LayerNorm(292) stats, 8 threads per node + shuffle reduce ----
    {
        int node = tid >> 3;          // 0..31
        int l8   = tid & 7;
        const float* xr = &s_x[node * DMSG];
        float s = 0.f, ss = 0.f;
        for (int k = l8; k < DMSG; k += 8) { float v = xr[k]; s += v; ss += v * v; }
        #pragma unroll
        for (int off = 4; off > 0; off >>= 1) {
            s  += __shfl_xor(s, off, 8);
            ss += __shfl_xor(ss, off, 8);
        }
        if (l8 == 0) {
            float mu = s * (1.0f / DMSG);
            float var = ss * (1.0f / DMSG) - mu * mu;
            s_mu[node] = mu;
            s_rstd[node] = rsqrtf(var + 1e-5f);
        }
    }
    __syncthreads();

    // ---- phase 4: normalize x in place ----
    for (int idx = tid; idx < NODES_PER_BLK * DMSG; idx += THREADS) {
        int node = idx / DMSG, k = idx % DMSG;
        s_x[idx] = (s_x[idx] - s_mu[node]) * s_rstd[node] * s_lng[k] + s_lnb[k];
    }
    __syncthreads();

    // ---- phase 5: WMMA GEMMs: gi = x @ W_ih^T, gh = mem @ W_hh^T ----
    float* s_acci = s_scr;                       // [32][300]
    float* s_acch = s_scr + NODES_PER_BLK * G3;  // [32][300]
    {
        const int NT = 19;                       // ceil(300/16)
        const int MT = NODES_PER_BLK / 16;
        for (int t = wave; t < MT * NT; t += (THREADS >> 5)) {
            int msub = t / NT, nt = t % NT;
            int row  = msub * 16 + l16;
            int ncol = nt * 16 + l16;
            int ncl  = ncol < G3 ? ncol : (G3 - 1);   // clamp B loads past N=300
            v8f ci = {}, ch = {};

            const float* wr = wih + (size_t)ncl * DMSG;
            const float* xr = &s_x[row * DMSG];
            for (int k0 = 0; k0 < DMSG; k0 += 4) {
                int kk = k0 + kh;
                v2f a = *(const v2f*)(xr + kk);       // A: 16x4 f32 fragment
                v2f b = *(const v2f*)(wr + kk);       // B[k][n] = W[n][k]
                ci = __builtin_amdgcn_wmma_f32_16x16x4_f32(
                        false, a, false, b, (short)0, ci, false, false);
            }
            const float* hr = whh + (size_t)ncl * MEMD;
            const float* mr = &s_mem[row * MEMD];
            for (int k0 = 0; k0 < MEMD; k0 += 4) {
                int kk = k0 + kh;
                v2f a = *(const v2f*)(mr + kk);
                v2f b = *(const v2f*)(hr + kk);
                ch = __builtin_amdgcn_wmma_f32_16x16x4_f32(
                        false, a, false, b, (short)0, ch, false, false);
            }

            if (ncol < G3) {
                int rbase = msub * 16 + rhalf;        // C: M = 8*(lane>=16)+v
                #pragma unroll
                for (int v = 0; v < 8; ++v) {
                    s_acci[(rbase + v) * G3 + ncol] = ci[v];
                    s_acch[(rbase + v) * G3 + ncol] = ch[v];
                }
            }
        }
    }
    __syncthreads();

    // ---- phase 6: GRU gates + output ----
    for (int idx = tid; idx < NODES_PER_BLK * MEMD; idx += THREADS) {
        int node = idx / MEMD, j = idx % MEMD;
        int g = base + node;
        if (g < Ntot) {
            const float* ai = &s_acci[node * G3];
            const float* ah = &s_acch[node * G3];
            float r  = sigmoidf_(ai[j] + ah[j]);
            float z  = sigmoidf_(ai[MEMD + j] + ah[MEMD + j]);
            float ng = tanhf(ai[2 * MEMD + j] + r * ah[2 * MEMD + j]);
            float hprev = s_mem[idx];
            out[(size_t)g * MEMD + j] = (1.0f - z) * ng + z * hprev;
        }
    }
}

extern "C" void kernel_launch(void* const* d_in, const int* in_sizes, int n_in,
                              void* d_out, int out_size, void* d_ws, size_t ws_size,
                              hipStream_t stream) {
    const int*   n_idp  = (const int*)d_in[0];
    const float* memory = (const float*)d_in[1];
    const float* msgd   = (const float*)d_in[2];
    const float* w1     = (const float*)d_in[3];
    const float* b1     = (const float*)d_in[4];
    const float* w2     = (const float*)d_in[5];
    const float* b2     = (const float*)d_in[6];
    const float* lg     = (const float*)d_in[7];
    const float* lb     = (const float*)d_in[8];
    const float* lng    = (const float*)d_in[9];
    const float* lnb    = (const float*)d_in[10];
    const float* wih    = (const float*)d_in[11];
    const float* whh    = (const float*)d_in[12];
    float* out = (float*)d_out;

    int Ntot   = in_sizes[0];
    int blocks = (Ntot + NODES_PER_BLK - 1) / NODES_PER_BLK;
    tgn_update_kernel<<<blocks, THREADS, 0, stream>>>(
        n_idp, memory, msgd, w1, b1, w2, b2, lg, lb, lng, lnb, wih, whh, out, Ntot);
}